// RSA_22213570855053
// MI455X (gfx1250) — compile-verified
//
#include <hip/hip_runtime.h>
#include <hip/hip_bf16.h>

// ---------------------------------------------------------------------------
// Gated linear-attention forward for MI455X (gfx1250), fp32 via V_WMMA_F32_16X16X4_F32
//   K1: h = input @ w_in                  (WMMA f32 GEMM, async-LDS double-buffered)
//   K2: fused conv4+SiLU+sigmoid + scan   (1 WG per (b,head), WMMA f32 64x64x64/step,
//                                          async-LDS prefetch of activation chunks)
//   K3: z = rmsnorm(y * silu(gate)) * g   (in-place on y)
//   K4: out = z @ w_out                   (WMMA f32 GEMM)
// Workspace: h (8192*5120 f32) + y (8192*1024 f32) = 192 MiB
// ---------------------------------------------------------------------------

typedef float v2f __attribute__((ext_vector_type(2)));
typedef float v8f __attribute__((ext_vector_type(8)));
typedef int   i32x4v __attribute__((vector_size(4 * sizeof(int))));

#if defined(__gfx1250__) && __has_builtin(__builtin_amdgcn_wmma_f32_16x16x4_f32)
#define HAVE_WMMA_F32X4 1
#else
#define HAVE_WMMA_F32X4 0
#endif

#if defined(__gfx1250__) &&                                                   \
    __has_builtin(__builtin_amdgcn_global_load_async_to_lds_b128) &&          \
    __has_builtin(__builtin_amdgcn_global_load_async_to_lds_b32)
#define HAVE_ASYNC_LDS 1
#else
#define HAVE_ASYNC_LDS 0
#endif

__device__ __forceinline__ v8f wmma_f32x4(v2f a, v2f b, v8f c) {
#if HAVE_WMMA_F32X4
  // 8 args: (neg_a, A, neg_b, B, c_mod, C, reuse_a, reuse_b)
  return __builtin_amdgcn_wmma_f32_16x16x4_f32(false, a, false, b, (short)0, c,
                                               false, false);
#else
  // compile-only fallback (host pass); not numerically meaningful
  #pragma unroll
  for (int e = 0; e < 8; ++e) c[e] += a[0] * b[0] + a[1] * b[1];
  return c;
#endif
}

// ---- async global -> LDS copy helpers (ASYNCcnt-tracked on gfx1250) -------
#if HAVE_ASYNC_LDS
typedef __attribute__((address_space(1))) i32x4v* g_v4p;
typedef __attribute__((address_space(3))) i32x4v* l_v4p;
typedef __attribute__((address_space(1))) int*    g_i32p;
typedef __attribute__((address_space(3))) int*    l_i32p;
__device__ __forceinline__ void async_b128(void* lds, const void* g) {
  __builtin_amdgcn_global_load_async_to_lds_b128((g_v4p)g, (l_v4p)lds, 0, 0);
}
__device__ __forceinline__ void async_b32(void* lds, const void* g) {
  __builtin_amdgcn_global_load_async_to_lds_b32((g_i32p)g, (l_i32p)lds, 0, 0);
}
__device__ __forceinline__ void wait_async() {
#if __has_builtin(__builtin_amdgcn_s_wait_asynccnt)
  __builtin_amdgcn_s_wait_asynccnt(0);
#else
  asm volatile("s_wait_asynccnt 0x0" ::: "memory");
#endif
}
#else
__device__ __forceinline__ void async_b128(void* lds, const void* g) {
  *(float4*)lds = *(const float4*)g;
}
__device__ __forceinline__ void async_b32(void* lds, const void* g) {
  *(float*)lds = *(const float*)g;
}
__device__ __forceinline__ void wait_async() {}
#endif

// ---------------------------------------------------------------------------
// K1/K4: fp32 WMMA GEMM.  C[M,N] = A[M,K] @ B[K,N].  M%128==0, N%256==0, K%16==0.
// 256 threads = 8 waves in 2(M)x4(N); each wave computes 64x64 (4x4 C tiles).
// Double-buffered LDS tiles filled by GLOBAL_LOAD_ASYNC_TO_LDS_B128.
// ---------------------------------------------------------------------------
#define GTM 128
#define GTN 256
#define GTK 16

__global__ __launch_bounds__(256, 1) void f32_wmma_gemm(
    const float* __restrict__ A, const float* __restrict__ B,
    float* __restrict__ C, int M, int N, int K) {
  // stride 20 floats: 16B-aligned rows for b128 async writes, and
  // 20*m mod 64 spans 16 distinct banks across the 16 fragment lanes.
  __shared__ float As[2][GTM][20];
  __shared__ float Bs[2][GTK][260];  // 260*4B = 16B-aligned row stride

  const int tid  = threadIdx.x;
  const int lane = tid & 31;
  const int wid  = tid >> 5;
  const int wm   = wid >> 2;            // 0..1
  const int wn   = wid & 3;             // 0..3
  const int m0   = blockIdx.y * GTM;
  const int n0   = blockIdx.x * GTN;
  const int lm   = lane & 15;           // col / row-within-tile
  const int kh   = (lane >> 4) * 2;     // K half-offset: 0 or 2
  const int mh   = (lane >> 4) * 8;     // C row half-offset: 0 or 8

  v8f acc[4][4];
  #pragma unroll
  for (int i = 0; i < 4; ++i)
    #pragma unroll
    for (int j = 0; j < 4; ++j)
      #pragma unroll
      for (int e = 0; e < 8; ++e) acc[i][j][e] = 0.0f;

  auto issue_tiles = [&](int kk, int buf) {
    // A tile: 128x16 -> 512 x b128, 2 per thread
    #pragma unroll
    for (int i = 0; i < 2; ++i) {
      const int e = i * 256 + tid;
      const int r = e >> 2, c4 = (e & 3) << 2;
      async_b128(&As[buf][r][c4], &A[(size_t)(m0 + r) * K + kk + c4]);
    }
    // B tile: 16x256 -> 1024 x b128, 4 per thread
    #pragma unroll
    for (int i = 0; i < 4; ++i) {
      const int e = i * 256 + tid;
      const int r = e >> 6, c4 = (e & 63) << 2;
      async_b128(&Bs[buf][r][c4], &B[(size_t)(kk + r) * N + n0 + c4]);
    }
  };

  issue_tiles(0, 0);
  const int niter = K / GTK;
  for (int it = 0; it < niter; ++it) {
    const int cur = it & 1;
    wait_async();        // my async LDS writes for tile `it` have landed
    __syncthreads();     // everyone's have; prev compute on buf cur^1 is done
    if (it + 1 < niter) issue_tiles((it + 1) * GTK, cur ^ 1);

    #pragma unroll
    for (int k4 = 0; k4 < 4; ++k4) {
      const int kb = k4 * 4 + kh;
      v2f afr[4], bfr[4];
      #pragma unroll
      for (int i = 0; i < 4; ++i) {
        const float* ap = &As[cur][wm * 64 + i * 16 + lm][kb];
        afr[i][0] = ap[0];
        afr[i][1] = ap[1];
      }
      #pragma unroll
      for (int j = 0; j < 4; ++j) {
        const int cc = wn * 64 + j * 16 + lm;
        bfr[j][0] = Bs[cur][kb][cc];
        bfr[j][1] = Bs[cur][kb + 1][cc];
      }
      #pragma unroll
      for (int i = 0; i < 4; ++i)
        #pragma unroll
        for (int j = 0; j < 4; ++j)
          acc[i][j] = wmma_f32x4(afr[i], bfr[j], acc[i][j]);
    }
  }

  // Write back: C vgpr r holds row r + mh, col lm within each 16x16 tile.
  #pragma unroll
  for (int i = 0; i < 4; ++i)
    #pragma unroll
    for (int j = 0; j < 4; ++j) {
      const int rbase = m0 + wm * 64 + i * 16 + mh;
      const int cc    = n0 + wn * 64 + j * 16 + lm;
      #pragma unroll
      for (int r = 0; r < 8; ++r)
        C[(size_t)(rbase + r) * N + cc] = acc[i][j][r];
    }
}

// ---------------------------------------------------------------------------
// K2: fused depthwise conv(K=4)+SiLU+sigmoid + linear-attention scan.
// One workgroup (256 thr = 8 waves) per (b, head); 32 workgroups total.
// State s [64x64] double-buffered in LDS; s@W via f32 WMMA (16 tiles, 2/wave).
// Activation chunks (16 steps + 3-row causal halo) prefetched asynchronously.
// ---------------------------------------------------------------------------
__global__ __launch_bounds__(256, 1) void scan_kernel(
    const float* __restrict__ h,       // [B*S][5120]
    const float* __restrict__ conv_w,  // [4096][4]
    const float* __restrict__ state_w, // [16][64][64]
    const float* __restrict__ Dm,      // [16][64]
    float* __restrict__ y) {           // [B*S][1024]
  const int bid  = blockIdx.x;
  const int b    = bid >> 4;
  const int head = bid & 15;
  const int tid  = threadIdx.x;
  const int lane = tid & 31;
  const int wid  = tid >> 5;
  const int lm   = lane & 15;
  const int kh   = (lane >> 4) * 2;
  const int mh   = (lane >> 4) * 8;

  __shared__ float Ws[64][68];
  __shared__ float sA[64][68];
  __shared__ float sB[64][68];
  __shared__ float hbuf[2][19][256];
  __shared__ float qv[64], kv[64], vv[64], fv[64], Dv[64];
  __shared__ float ypart[4][64];

  const int g  = tid >> 6;                       // 0=q 1=k 2=v 3=f
  const int c  = tid & 63;
  const int ch = g * 1024 + head * 64 + c;       // channel in conv region of h
  const float4 wc = *(const float4*)&conv_w[ch * 4];

  for (int i = tid; i < 64 * 64; i += 256) {
    Ws[i >> 6][i & 63] = state_w[head * 4096 + i];
    sA[i >> 6][i & 63] = 0.0f;
  }
  if (tid < 64) Dv[tid] = Dm[head * 64 + tid];

  float (*scur)[68] = sA;
  float (*snxt)[68] = sB;

  const float* hb = h + (size_t)b * 4096 * 5120;
  float*       yb = y + (size_t)b * 4096 * 1024 + head * 64;

  auto issue_chunk = [&](int t0, int buf) {
    // rows t0-3 .. t0+15 of this head's 256 channels
    #pragma unroll 4
    for (int r = 0; r < 19; ++r) {
      const int t = t0 - 3 + r;
      if (t >= 0)
        async_b32(&hbuf[buf][r][tid], &hb[(size_t)t * 5120 + ch]);
      else
        hbuf[buf][r][tid] = 0.0f;  // causal zero pad (first chunk only)
    }
  };

  issue_chunk(0, 0);
  for (int t0 = 0, cidx = 0; t0 < 4096; t0 += 16, ++cidx) {
    const int cur = cidx & 1;
    wait_async();
    __syncthreads();  // chunk `cidx` resident; prev chunk fully consumed
    if (t0 + 16 < 4096) issue_chunk(t0 + 16, cur ^ 1);

    for (int r = 0; r < 16; ++r) {
      // depthwise causal conv + activation -> q,k,v (SiLU), f (sigmoid)
      {
        const float x = hbuf[cur][r][tid] * wc.x + hbuf[cur][r + 1][tid] * wc.y +
                        hbuf[cur][r + 2][tid] * wc.z + hbuf[cur][r + 3][tid] * wc.w;
        const float sg = 1.0f / (1.0f + __expf(-x));
        const float val = (g == 3) ? sg : x * sg;
        if (g == 0)      qv[c] = val;
        else if (g == 1) kv[c] = val;
        else if (g == 2) vv[c] = val;
        else             fv[c] = val;
      }
      __syncthreads();

      // s_new = f * (s @ W) + k (x) v    -- 16 tiles of 16x16, 2 per wave
      #pragma unroll
      for (int tp = 0; tp < 2; ++tp) {
        const int tile = wid * 2 + tp;
        const int ti = tile >> 2, tj = tile & 3;
        v8f accv;
        #pragma unroll
        for (int e = 0; e < 8; ++e) accv[e] = 0.0f;
        #pragma unroll
        for (int k4 = 0; k4 < 16; ++k4) {
          const int kb = k4 * 4 + kh;
          v2f a, bf;
          a[0]  = scur[ti * 16 + lm][kb];
          a[1]  = scur[ti * 16 + lm][kb + 1];
          bf[0] = Ws[kb][tj * 16 + lm];
          bf[1] = Ws[kb + 1][tj * 16 + lm];
          accv = wmma_f32x4(a, bf, accv);
        }
        #pragma unroll
        for (int e = 0; e < 8; ++e) {
          const int row = ti * 16 + e + mh;
          const int col = tj * 16 + lm;
          snxt[row][col] = fv[row] * accv[e] + kv[row] * vv[col];
        }
      }
      __syncthreads();

      // y_t = q . s_new  (+ residual v*D), 4-way split over KD
      {
        const int vd = tid & 63, part = tid >> 6;
        float a = 0.0f;
        #pragma unroll
        for (int k = 0; k < 16; ++k)
          a += qv[part * 16 + k] * snxt[part * 16 + k][vd];
        ypart[part][vd] = a;
      }
      __syncthreads();
      if (tid < 64) {
        const float yy = ypart[0][tid] + ypart[1][tid] + ypart[2][tid] +
                         ypart[3][tid] + vv[tid] * Dv[tid];
        yb[(size_t)(t0 + r) * 1024 + tid] = yy;
      }
      float (*tmp)[68] = scur; scur = snxt; snxt = tmp;
      __syncthreads();
    }
  }
}

// ---------------------------------------------------------------------------
// K3: z = rmsnorm(y * silu(gate)) * gamma, in-place on y.  One WG per row.
// ---------------------------------------------------------------------------
__global__ __launch_bounds__(256, 4) void norm_kernel(
    float* __restrict__ y,           // [8192][1024] in/out
    const float* __restrict__ h,     // [8192][5120]; gate at cols [4096,5120)
    const float* __restrict__ gamma) // [1024]
{
  __shared__ float red[8];
  __shared__ float sscale;
  const int row = blockIdx.x;
  const int tid = threadIdx.x;
  float* yr       = y + (size_t)row * 1024;
  const float* gr = h + (size_t)row * 5120 + 4096;

  float z[4];
  float ss = 0.0f;
  #pragma unroll
  for (int i = 0; i < 4; ++i) {
    const int idx = tid + i * 256;
    const float gv = gr[idx];
    const float zz = yr[idx] * (gv / (1.0f + __expf(-gv)));
    z[i] = zz;
    ss += zz * zz;
  }
  #pragma unroll
  for (int off = 16; off > 0; off >>= 1) ss += __shfl_down(ss, off, 32);
  if ((tid & 31) == 0) red[tid >> 5] = ss;
  __syncthreads();
  if (tid == 0) {
    float tot = 0.0f;
    #pragma unroll
    for (int i = 0; i < 8; ++i) tot += red[i];
    sscale = rsqrtf(tot * (1.0f / 1024.0f) + 1e-6f);
  }
  __syncthreads();
  const float sc = sscale;
  #pragma unroll
  for (int i = 0; i < 4; ++i) {
    const int idx = tid + i * 256;
    yr[idx] = z[i] * sc * gamma[idx];
  }
}

// ---------------------------------------------------------------------------
extern "C" void kernel_launch(void* const* d_in, const int* in_sizes, int n_in,
                              void* d_out, int out_size, void* d_ws, size_t ws_size,
                              hipStream_t stream) {
  const float* input   = (const float*)d_in[0]; // [2,4096,2048]
  const float* w_in    = (const float*)d_in[1]; // [2048,5120]
  const float* conv_w  = (const float*)d_in[2]; // [4096,1,4]
  const float* state_w = (const float*)d_in[3]; // [16,64,64]
  const float* Dm      = (const float*)d_in[4]; // [16,64]
  const float* gamma   = (const float*)d_in[5]; // [1024]
  const float* w_out   = (const float*)d_in[6]; // [1024,2048]
  float* out = (float*)d_out;

  float* h = (float*)d_ws;                      // 8192*5120 f32
  float* y = h + (size_t)8192 * 5120;           // 8192*1024 f32

  dim3 blk(256);
  // K1: h = input @ w_in    (M=8192, N=5120, K=2048)
  f32_wmma_gemm<<<dim3(5120 / GTN, 8192 / GTM), blk, 0, stream>>>(
      input, w_in, h, 8192, 5120, 2048);
  // K2: fused conv + scan -> y
  scan_kernel<<<dim3(32), blk, 0, stream>>>(h, conv_w, state_w, Dm, y);
  // K3: y <- rmsnorm(y * silu(gate)) * gamma
  norm_kernel<<<dim3(8192), blk, 0, stream>>>(y, h, gamma);
  // K4: out = y @ w_out     (M=8192, N=2048, K=1024)
  f32_wmma_gemm<<<dim3(2048 / GTN, 8192 / GTM), blk, 0, stream>>>(
      y, w_out, out, 8192, 2048, 1024);
}